// OmniNet_13408887898469
// MI455X (gfx1250) — compile-verified
//
#include <hip/hip_runtime.h>

// OmniNet (SchNet v1, 2 interactions) fused single-kernel implementation for
// gfx1250. One wave32 per batch element; w_net tiny-MLP GEMMs run on
// V_WMMA_F32_16X16X4_F32 (K=4 == dist_feat_dim, full K utilization).
// A-tile global loads are software double-buffered across loop iterations.

#define NE 32   // electrons
#define NA 8    // atoms
#define DF 4    // dist features
#define KD 8    // kernel dim
#define ED 16   // embedding dim
#define NI 2    // interactions

#define HSTRIDE 36   // padded j-stride for sh_h rows (bank-conflict-free, 16B-aligned)
#define SSTRIDE 18   // padded col-stride for stage rows (bank-conflict-free, 8B-aligned)

typedef float v2f __attribute__((ext_vector_type(2)));
typedef float v4f __attribute__((ext_vector_type(4)));
typedef float v8f __attribute__((ext_vector_type(8)));

// softplus(x) - log(2) = ln2 * log2(1 + 2^(x*log2e)) - ln2
// Raw v_exp_f32 / v_log_f32 (TRANS ops co-execute with the WMMA pipe).
__device__ __forceinline__ float ssp_fast(float x) {
    float t = __builtin_amdgcn_exp2f(x * 1.4426950408889634f);
    float l = __builtin_amdgcn_logf(1.0f + t);
    return fmaf(l, 0.6931471805599453f, -0.6931471805599453f);
}

__global__ __launch_bounds__(32) void omninet_kernel(
    const float* __restrict__ dn,   // (B,32,8,4)
    const float* __restrict__ de,   // (B,32,32,4)
    const float* __restrict__ X,    // (1,16)
    const float* __restrict__ Y,    // (8,8)
    const float* __restrict__ w1,   // (2,4,6)
    const float* __restrict__ wb1,  // (2,6)
    const float* __restrict__ w2,   // (2,6,8)
    const float* __restrict__ wb2,  // (2,8)
    const float* __restrict__ hsw,  // (2,16,8)
    const float* __restrict__ hsb,  // (2,8)
    const float* __restrict__ haw,  // (2,16,8)
    const float* __restrict__ hab,  // (2,8)
    const float* __restrict__ gw,   // (2,8,16)
    const float* __restrict__ gb,   // (2,16)
    const float* __restrict__ orbw, // (16)
    float* __restrict__ out)        // (B)
{
    __shared__ float sh_x[NE][ED];              // electron embeddings
    __shared__ float sh_h[2][16][HSTRIDE];      // TRANSPOSED: [spin][k][j]; k rows 8..15 zero
    __shared__ float sh_z[NE][KD];              // messages
    __shared__ float sh_stage[16][SSTRIDE];     // layer1->layer2 transpose staging

    const int b    = blockIdx.x;
    const int L    = threadIdx.x;   // 0..31
    const int col  = L & 15;        // N index (k / hidden col)
    const int half = L >> 4;        // lane half (K split for A/B packing)

    // initial embeddings: all electrons share X[0]
    #pragma unroll
    for (int e = 0; e < ED; ++e) sh_x[L][e] = X[e];
    // zero padded H rows (k = 8..15) so pad lanes read defined data
    #pragma unroll
    for (int k = 8; k < 16; ++k) { sh_h[0][k][L] = 0.0f; sh_h[1][k][L] = 0.0f; }

    // per-lane Y column: Yk[m] = Y[m][col]
    float Yk[NA];
    #pragma unroll
    for (int m = 0; m < NA; ++m) Yk[m] = (col < KD) ? Y[m * KD + col] : 0.0f;

    const float* deb = de + (size_t)b * NE * NE * DF;
    const float* dnb = dn + (size_t)b * NE * NA * DF;

    for (int n = 0; n < NI; ++n) {
        // ---- pack w_net layer-1 B (4x16, cols>=6 zero) + bias C ----
        // B layout (32-bit, KxN): VGPR0 = rows {0 | 2} across lane halves, VGPR1 = {1 | 3}
        v2f b1;
        b1.x = (col < 6) ? w1[n * 24 + (half ? 2 : 0) * 6 + col] : 0.0f;
        b1.y = (col < 6) ? w1[n * 24 + (half ? 3 : 1) * 6 + col] : 0.0f;
        float bias1 = (col < 6) ? wb1[n * 6 + col] : 0.0f;
        v8f c1;
        #pragma unroll
        for (int v = 0; v < 8; ++v) c1[v] = bias1;

        // ---- pack w_net layer-2 B as two K=4 slabs: rows 0..3 and rows 4,5 (+zeros) ----
        v2f b2lo, b2hi;
        b2lo.x = (col < KD) ? w2[n * 48 + (half ? 2 : 0) * KD + col] : 0.0f;
        b2lo.y = (col < KD) ? w2[n * 48 + (half ? 3 : 1) * KD + col] : 0.0f;
        b2hi.x = (col < KD && !half) ? w2[n * 48 + 4 * KD + col] : 0.0f; // rows 6,7 pad -> 0
        b2hi.y = (col < KD && !half) ? w2[n * 48 + 5 * KD + col] : 0.0f;
        float bias2 = (col < KD) ? wb2[n * KD + col] : 0.0f;
        v8f c2b;
        #pragma unroll
        for (int v = 0; v < 8; ++v) c2b[v] = bias2;

        // ---- spin message features hs/ha (transposed store [spin][k][j]): lane = j ----
        {
            const int j = L;
            #pragma unroll
            for (int k = 0; k < KD; ++k) {
                float s = hsb[n * KD + k];
                float a = hab[n * KD + k];
                #pragma unroll
                for (int e = 0; e < ED; ++e) {
                    float xv = sh_x[j][e];
                    s = fmaf(xv, hsw[n * ED * KD + e * KD + k], s);
                    a = fmaf(xv, haw[n * ED * KD + e * KD + k], a);
                }
                sh_h[0][k][j] = s;
                sh_h[1][k][j] = a;
            }
        }

        // ---- electron-electron messages: z[i,k] = sum_j mask * we[i,j,k] * h[j,k] ----
        // Software double-buffer the A fragments (hide global/L2 latency behind
        // a full chunk of WMMA + SSP + LDS work).
        v2f aCur[2];
        #pragma unroll
        for (int jb = 0; jb < 2; ++jb) {
            const float* s0 = deb + (size_t)(jb * 16 + col) * DF + 2 * half;
            aCur[jb].x = s0[0]; aCur[jb].y = s0[1];
        }
        for (int i = 0; i < NE; ++i) {
            v2f aNext[2];
            const int ip = (i + 1) & 31;   // wraps to row 0 on last iter (always valid)
            #pragma unroll
            for (int jb = 0; jb < 2; ++jb) {
                const float* s0 = deb + (size_t)(ip * NE + jb * 16 + col) * DF + 2 * half;
                aNext[jb].x = s0[0]; aNext[jb].y = s0[1];
            }
            float acc = 0.0f;
            #pragma unroll
            for (int jb = 0; jb < 2; ++jb) {
                const int j0 = jb * 16;
                v8f d1 = __builtin_amdgcn_wmma_f32_16x16x4_f32(
                    false, aCur[jb], false, b1, (short)0, c1, false, false);
                // SSP + stage (only cols 0..7 are ever re-read as A')
                if (col < KD) {
                    #pragma unroll
                    for (int v = 0; v < 8; ++v)
                        sh_stage[v + 8 * half][col] = ssp_fast(d1[v]);
                }
                v2f alo = *(const v2f*)&sh_stage[col][2 * half];
                v2f ahi = *(const v2f*)&sh_stage[col][4 + 2 * half];
                v8f d2 = __builtin_amdgcn_wmma_f32_16x16x4_f32(
                    false, alo, false, b2lo, (short)0, c2b, false, false);
                d2 = __builtin_amdgcn_wmma_f32_16x16x4_f32(
                    false, ahi, false, b2hi, (short)0, d2, false, false);
                // contract this chunk against spin-selected h (vector LDS loads)
                const int hsel = ((i >> 4) == jb) ? 0 : 1;
                const float* hrow = &sh_h[hsel][col][j0 + 8 * half];
                v4f h0 = *(const v4f*)hrow;
                v4f h1 = *(const v4f*)(hrow + 4);
                if ((i >> 4) == jb) {   // uniform: this chunk contains the diagonal
                    const int idx = i - (j0 + 8 * half);  // lane-dependent, zero elem idx
                    h0.x = (idx == 0) ? 0.0f : h0.x;
                    h0.y = (idx == 1) ? 0.0f : h0.y;
                    h0.z = (idx == 2) ? 0.0f : h0.z;
                    h0.w = (idx == 3) ? 0.0f : h0.w;
                    h1.x = (idx == 4) ? 0.0f : h1.x;
                    h1.y = (idx == 5) ? 0.0f : h1.y;
                    h1.z = (idx == 6) ? 0.0f : h1.z;
                    h1.w = (idx == 7) ? 0.0f : h1.w;
                }
                acc = fmaf(d2[0], h0.x, acc);
                acc = fmaf(d2[1], h0.y, acc);
                acc = fmaf(d2[2], h0.z, acc);
                acc = fmaf(d2[3], h0.w, acc);
                acc = fmaf(d2[4], h1.x, acc);
                acc = fmaf(d2[5], h1.y, acc);
                acc = fmaf(d2[6], h1.z, acc);
                acc = fmaf(d2[7], h1.w, acc);
            }
            // fold the two lane halves (same k, disjoint j sets)
            float tot = acc + __shfl_xor(acc, 16, 32);
            if (L < KD) sh_z[i][L] = tot;
            aCur[0] = aNext[0];
            aCur[1] = aNext[1];
        }

        // ---- nuclear messages: rows (i,m) flattened; chunk c covers i = 2c, 2c+1 ----
        v2f aNuc;
        {
            const float* s0 = dnb + (size_t)col * DF + 2 * half;
            aNuc.x = s0[0]; aNuc.y = s0[1];
        }
        for (int c = 0; c < 16; ++c) {
            v2f aNucNext;
            const int cp = (c + 1) & 15;
            {
                const float* s0 = dnb + (size_t)(cp * 16 + col) * DF + 2 * half;
                aNucNext.x = s0[0]; aNucNext.y = s0[1];
            }
            v8f d1 = __builtin_amdgcn_wmma_f32_16x16x4_f32(
                false, aNuc, false, b1, (short)0, c1, false, false);
            if (col < KD) {
                #pragma unroll
                for (int v = 0; v < 8; ++v)
                    sh_stage[v + 8 * half][col] = ssp_fast(d1[v]);
            }
            v2f alo = *(const v2f*)&sh_stage[col][2 * half];
            v2f ahi = *(const v2f*)&sh_stage[col][4 + 2 * half];
            v8f d2 = __builtin_amdgcn_wmma_f32_16x16x4_f32(
                false, alo, false, b2lo, (short)0, c2b, false, false);
            d2 = __builtin_amdgcn_wmma_f32_16x16x4_f32(
                false, ahi, false, b2hi, (short)0, d2, false, false);
            // lane half 0 -> electron 2c (m = v), half 1 -> electron 2c+1 (m = v)
            float acc = 0.0f;
            #pragma unroll
            for (int v = 0; v < 8; ++v) acc = fmaf(d2[v], Yk[v], acc);
            if (col < KD) sh_z[2 * c + half][col] += acc;
            aNuc = aNucNext;
        }

        // ---- residual embedding update: x[i] += z[i] @ g_w + g_b ----
        {
            const int i = L;
            float zi[KD];
            #pragma unroll
            for (int k = 0; k < KD; ++k) zi[k] = sh_z[i][k];
            #pragma unroll
            for (int e = 0; e < ED; ++e) {
                float acc = gb[n * ED + e];
                #pragma unroll
                for (int k = 0; k < KD; ++k)
                    acc = fmaf(zi[k], gw[n * KD * ED + k * ED + e], acc);
                sh_x[i][e] += acc;
            }
        }
    }

    // ---- orbital projection + sum over electrons ----
    float p = 0.0f;
    #pragma unroll
    for (int e = 0; e < ED; ++e) p = fmaf(sh_x[L][e], orbw[e], p);
    p += __shfl_xor(p, 1, 32);
    p += __shfl_xor(p, 2, 32);
    p += __shfl_xor(p, 4, 32);
    p += __shfl_xor(p, 8, 32);
    p += __shfl_xor(p, 16, 32);
    if (L == 0) out[b] = p;
}

extern "C" void kernel_launch(void* const* d_in, const int* in_sizes, int n_in,
                              void* d_out, int out_size, void* d_ws, size_t ws_size,
                              hipStream_t stream) {
    (void)n_in; (void)d_ws; (void)ws_size; (void)out_size;
    const int B = in_sizes[0] / (NE * NA * DF);  // dists_nuc is (B,32,8,4)
    omninet_kernel<<<B, 32, 0, stream>>>(
        (const float*)d_in[0],  // dists_nuc
        (const float*)d_in[1],  // dists_elec
        (const float*)d_in[2],  // X
        (const float*)d_in[3],  // Y
        (const float*)d_in[4],  // w1
        (const float*)d_in[5],  // wb1
        (const float*)d_in[6],  // w2
        (const float*)d_in[7],  // wb2
        (const float*)d_in[8],  // hs_w
        (const float*)d_in[9],  // hs_b
        (const float*)d_in[10], // ha_w
        (const float*)d_in[11], // ha_b
        (const float*)d_in[12], // g_w
        (const float*)d_in[13], // g_b
        (const float*)d_in[14], // orb_w
        (float*)d_out);
}